// Model_85323820302368
// MI455X (gfx1250) — compile-verified
//
#include <hip/hip_runtime.h>
#include <hip/hip_bf16.h>

typedef float v2f __attribute__((ext_vector_type(2)));
typedef float v8f __attribute__((ext_vector_type(8)));

// ---------------------------------------------------------------------------
// Degree / dinv kernels
// ---------------------------------------------------------------------------
__global__ __launch_bounds__(256) void deg_init_kernel(float* __restrict__ deg, int N) {
    int i = blockIdx.x * blockDim.x + threadIdx.x;
    if (i < N) deg[i] = 1.0f;  // self loop
}

__global__ __launch_bounds__(256) void deg_count_kernel(float* __restrict__ deg,
                                                        const int* __restrict__ dst, int E) {
    int e = blockIdx.x * blockDim.x + threadIdx.x;
    if (e < E) unsafeAtomicAdd(&deg[dst[e]], 1.0f);
}

__global__ __launch_bounds__(256) void deg_inv_kernel(float* __restrict__ deg, int N) {
    int i = blockIdx.x * blockDim.x + threadIdx.x;
    if (i < N) deg[i] = rsqrtf(fmaxf(deg[i], 1.0f));
}

// ---------------------------------------------------------------------------
// WMMA GEMM:  HS = dinv ⊙ (X @ W),  AGG initialized to HS (self-loop term).
// Block = 256 threads = 8 waves; wave w computes rows [bx*128 + w*16, +16),
// cols [by*16, +16). K = 256 for both layers.
// fp32 WMMA (v_wmma_f32_16x16x4_f32): A 16x4 (2 VGPR), B 4x16 (2 VGPR), C 16x16 (8 VGPR).
// ---------------------------------------------------------------------------
__global__ __launch_bounds__(256)
void gcn_gemm_kernel(const float* __restrict__ X, const float* __restrict__ W,
                     const float* __restrict__ dinv,
                     float* __restrict__ HS, float* __restrict__ AGG,
                     int N, int K, int D) {
    __shared__ float BsT[16 * 256];  // W panel transposed: BsT[n][k], 16KB

    const int tid  = threadIdx.x;
    const int wave = tid >> 5;
    const int lane = tid & 31;
    const int half = lane >> 4;   // 0: K=k0..k0+1, 1: K=k0+2..k0+3
    const int l15  = lane & 15;

    const int n_base = blockIdx.y * 16;
    const int m_base = blockIdx.x * 128 + wave * 16;

    // Stage W[k][n_base..n_base+15] transposed into LDS: thread t handles k=t.
    if (tid < K) {
        const float* wrow = W + (size_t)tid * D + n_base;
        #pragma unroll
        for (int j = 0; j < 16; ++j) BsT[j * 256 + tid] = wrow[j];
    }
    __syncthreads();

    // A fragment row for this lane (M = lane&15 in both halves); clamp for safe loads.
    int row  = m_base + l15;
    int rowc = row < N ? row : N - 1;
    const float* Arow = X + (size_t)rowc * K;
    const float* Brow = &BsT[l15 * 256];  // n = lane&15 for B/C/D striping

    v8f acc = (v8f)0.0f;
    #pragma unroll 8
    for (int k0 = 0; k0 < K; k0 += 4) {
        const int ka = k0 + half * 2;
        v2f a = *(const v2f*)(Arow + ka);   // A[m][ka], A[m][ka+1]
        v2f b = *(const v2f*)(Brow + ka);   // W[ka][n], W[ka+1][n]  (transposed LDS)
        acc = __builtin_amdgcn_wmma_f32_16x16x4_f32(false, a, false, b,
                                                    (short)0, acc, false, false);
    }

    // C/D layout: VGPR i -> row m_base + i + half*8, col n_base + (lane&15).
    #pragma unroll
    for (int i = 0; i < 8; ++i) {
        int r = m_base + i + half * 8;
        if (r < N) {
            float hv = acc[i] * dinv[r];
            size_t off = (size_t)r * D + n_base + l15;
            HS[off]  = hv;
            AGG[off] = hv;  // self-loop contribution pre-seeded
        }
    }
}

// ---------------------------------------------------------------------------
// Edge scatter:  AGG[dst[e]] += HS[src[e]]   (float4 per thread, fp32 atomics)
// eshift = log2(D/4): 6 for D=256, 5 for D=128.
// ---------------------------------------------------------------------------
__global__ __launch_bounds__(256)
void gcn_scatter_kernel(const float* __restrict__ HS, const int* __restrict__ src,
                        const int* __restrict__ dst, float* __restrict__ AGG,
                        int E, int D, int eshift) {
    long long t = (long long)blockIdx.x * blockDim.x + threadIdx.x;
    int e = (int)(t >> eshift);
    if (e >= E) return;
    int j = ((int)t & ((1 << eshift) - 1)) << 2;

    int s = src[e];
    int d = dst[e];
    const float4 v = *(const float4*)(HS + (size_t)s * D + j);
    float* p = AGG + (size_t)d * D + j;
    unsafeAtomicAdd(p + 0, v.x);
    unsafeAtomicAdd(p + 1, v.y);
    unsafeAtomicAdd(p + 2, v.z);
    unsafeAtomicAdd(p + 3, v.w);
}

// ---------------------------------------------------------------------------
// Finish:  OUT = dinv ⊙ AGG + b  (optional ReLU), float4, in-place safe.
// ---------------------------------------------------------------------------
__global__ __launch_bounds__(256)
void gcn_finish_kernel(float* __restrict__ OUT, const float* __restrict__ AGG,
                       const float* __restrict__ dinv, const float* __restrict__ bias,
                       int N, int D, int do_relu) {
    long long t = (long long)blockIdx.x * blockDim.x + threadIdx.x;
    long long total = (long long)N * (D >> 2);
    if (t >= total) return;
    int dq  = D >> 2;
    int row = (int)(t / dq);
    int c   = ((int)(t % dq)) << 2;

    float dv = dinv[row];
    const float4 a = *(const float4*)(AGG + (size_t)row * D + c);
    const float4 b = *(const float4*)(bias + c);
    float4 o;
    o.x = fmaf(dv, a.x, b.x);
    o.y = fmaf(dv, a.y, b.y);
    o.z = fmaf(dv, a.z, b.z);
    o.w = fmaf(dv, a.w, b.w);
    if (do_relu) {
        o.x = fmaxf(o.x, 0.0f); o.y = fmaxf(o.y, 0.0f);
        o.z = fmaxf(o.z, 0.0f); o.w = fmaxf(o.w, 0.0f);
    }
    *(float4*)(OUT + (size_t)row * D + c) = o;
}

// ---------------------------------------------------------------------------
// Launcher
// ---------------------------------------------------------------------------
extern "C" void kernel_launch(void* const* d_in, const int* in_sizes, int n_in,
                              void* d_out, int out_size, void* d_ws, size_t ws_size,
                              hipStream_t stream) {
    const int IN = 256, HID = 256, OUT = 128;
    const int N = in_sizes[0] / IN;
    const int E = in_sizes[1] / 2;

    const float* W1 = (const float*)d_in[4];
    const float* b1 = (const float*)d_in[5];
    const float* W2 = (const float*)d_in[6];
    const float* b2 = (const float*)d_in[7];

    char* ws = (char*)d_ws;
    float* dinv = (float*)ws;                              // N floats (16B-aligned: N*4 % 16 == 0)
    float* buf0 = (float*)(ws + (size_t)N * sizeof(float));  // N*256: HS (layer1) / HS2 (layer2)
    float* buf1 = buf0 + (size_t)N * HID;                    // N*256: AGG1 -> OUT1 in place
    float* out  = (float*)d_out;

    for (int g = 0; g < 2; ++g) {
        const float* x   = (const float*)d_in[g * 2];
        const int*   src = (const int*)d_in[g * 2 + 1];
        const int*   dst = src + E;

        // --- degrees -> dinv ---
        deg_init_kernel<<<(N + 255) / 256, 256, 0, stream>>>(dinv, N);
        deg_count_kernel<<<(E + 255) / 256, 256, 0, stream>>>(dinv, dst, E);
        deg_inv_kernel<<<(N + 255) / 256, 256, 0, stream>>>(dinv, N);

        // --- layer 1: hs = dinv*(x@W1); agg seeded with hs ---
        dim3 g1((N + 127) / 128, HID / 16);
        gcn_gemm_kernel<<<g1, 256, 0, stream>>>(x, W1, dinv, buf0, buf1, N, IN, HID);

        long long t1 = (long long)E * (HID / 4);
        gcn_scatter_kernel<<<(unsigned)((t1 + 255) / 256), 256, 0, stream>>>(
            buf0, src, dst, buf1, E, HID, 6);

        long long f1 = (long long)N * (HID / 4);
        gcn_finish_kernel<<<(unsigned)((f1 + 255) / 256), 256, 0, stream>>>(
            buf1, buf1, dinv, b1, N, HID, 1);

        // --- layer 2: hs2 = dinv*(out1@W2); agg seeded directly in d_out ---
        float* o = out + (size_t)g * N * OUT;
        dim3 g2((N + 127) / 128, OUT / 16);
        gcn_gemm_kernel<<<g2, 256, 0, stream>>>(buf1, W2, dinv, buf0, o, N, HID, OUT);

        long long t2 = (long long)E * (OUT / 4);
        gcn_scatter_kernel<<<(unsigned)((t2 + 255) / 256), 256, 0, stream>>>(
            buf0, src, dst, o, E, OUT, 5);

        long long f2 = (long long)N * (OUT / 4);
        gcn_finish_kernel<<<(unsigned)((f2 + 255) / 256), 256, 0, stream>>>(
            o, o, dinv, b2, N, OUT, 0);
    }
}